// ConceptWhiteningLayer_77137612636747
// MI455X (gfx1250) — compile-verified
//
#include <hip/hip_runtime.h>
#include <hip/hip_bf16.h>

// ---------------------------------------------------------------------------
// Concept Whitening forward for MI455X (gfx1250, wave32, WMMA).
//   out = (rot @ Sigma^{-1/2}) @ (x - mean)   fused as   M@x - M@mean
// Big GEMMs (Gram 128xM, apply 128x128 @ 128xM) via v_wmma_f32_16x16x4_f32.
// ---------------------------------------------------------------------------

typedef __attribute__((ext_vector_type(2))) float v2f;
typedef __attribute__((ext_vector_type(8))) float v8f;

constexpr int CH    = 128;        // channels
constexpr int BATCH = 64;         // batch
constexpr int HW    = 56 * 56;    // 3136 spatial positions
constexpr float MTOT = (float)BATCH * (float)HW;   // 200704 samples
constexpr float EPSR = 1e-5f;
constexpr int   T_NEWTON = 5;

// ---------------------------------------------------------------------------
// Kernel 1a: per (b, c) slab partial sums for the channel mean.
// ---------------------------------------------------------------------------
__global__ void __launch_bounds__(256)
cw_mean_partial(const float* __restrict__ x, float* __restrict__ meanPart) {
  const int b = blockIdx.x;
  const int c = blockIdx.y;
  const float* p = x + ((size_t)(b * CH + c)) * HW;
  float s = 0.f;
  for (int i = threadIdx.x; i < HW; i += 256) s += p[i];
  __shared__ float red[256];
  red[threadIdx.x] = s;
  __syncthreads();
  for (int off = 128; off > 0; off >>= 1) {
    if ((int)threadIdx.x < off) red[threadIdx.x] += red[threadIdx.x + off];
    __syncthreads();
  }
  if (threadIdx.x == 0) meanPart[c * BATCH + b] = red[0];
}

// ---------------------------------------------------------------------------
// Kernel 1b: finish mean (deterministic fixed-order reduction).
// ---------------------------------------------------------------------------
__global__ void cw_mean_finish(const float* __restrict__ meanPart,
                               float* __restrict__ mean) {
  const int c = threadIdx.x;  // 128 threads
  float s = 0.f;
  for (int b = 0; b < BATCH; ++b) s += meanPart[c * BATCH + b];
  mean[c] = s / MTOT;
}

// ---------------------------------------------------------------------------
// Kernel 2: per-batch-slab Gram partial  G_b = X_b X_b^T  (128 x 128).
// 8 waves; wave w owns output tile-rows [16w, 16w+16), 8 tiles of 16x16.
// K staged through LDS in chunks of 64.
// ---------------------------------------------------------------------------
constexpr int KC = 64;
constexpr int XPAD = KC + 4;   // pad so 3136-stride-ish LDS reads avoid bank conflicts

__global__ void __launch_bounds__(256)
cw_gram(const float* __restrict__ x, float* __restrict__ gramPart) {
  __shared__ float sX[CH][XPAD];       // 128 * 68 * 4B = 34.8 KB
  const int b    = blockIdx.x;
  const int tid  = threadIdx.x;
  const int wave = tid >> 5;
  const int lane = tid & 31;
  const int hi   = lane >> 4;          // lane-half: K offset 0 or 2
  const int l15  = lane & 15;

  v8f acc[8] = {};
  const float* xb = x + (size_t)b * CH * HW;

  for (int kc = 0; kc < HW; kc += KC) {
    // Stage 128 x 64 slab (2048 float4, 8 per thread).
#pragma unroll
    for (int i = 0; i < 8; ++i) {
      const int idx = i * 256 + tid;
      const int c   = idx >> 4;
      const int k4  = (idx & 15) << 2;
      const float4 v = *reinterpret_cast<const float4*>(xb + (size_t)c * HW + kc + k4);
      sX[c][k4 + 0] = v.x; sX[c][k4 + 1] = v.y;
      sX[c][k4 + 2] = v.z; sX[c][k4 + 3] = v.w;
    }
    __syncthreads();

#pragma unroll 2
    for (int k0 = 0; k0 < KC; k0 += 4) {
      const int ks = k0 + hi * 2;
      const int ra = wave * 16 + l15;
      v2f a; a.x = sX[ra][ks]; a.y = sX[ra][ks + 1];
#pragma unroll
      for (int tc = 0; tc < 8; ++tc) {
        const int rb = tc * 16 + l15;
        v2f bb; bb.x = sX[rb][ks]; bb.y = sX[rb][ks + 1];
        acc[tc] = __builtin_amdgcn_wmma_f32_16x16x4_f32(
            false, a, false, bb, (short)0, acc[tc], false, false);
      }
    }
    __syncthreads();
  }

  // Write 128x128 partial Gram (C/D layout: VGPR r -> M = r (+8 on hi half)).
  float* gp = gramPart + (size_t)b * CH * CH;
#pragma unroll
  for (int tc = 0; tc < 8; ++tc) {
#pragma unroll
    for (int r = 0; r < 8; ++r) {
      const int row = wave * 16 + r + hi * 8;
      const int col = tc * 16 + l15;
      gp[row * CH + col] = acc[tc][r];
    }
  }
}

// ---------------------------------------------------------------------------
// 128x128 matmul helper for the single-workgroup Newton kernel.
// 256 threads: thread t computes row (t>>1), 64-column half (t&1).
// ---------------------------------------------------------------------------
__device__ inline void mm128(float* __restrict__ Cm, const float* __restrict__ A,
                             const float* __restrict__ Bm, int tid) {
  const int row = tid >> 1;
  const int c0  = (tid & 1) << 6;
  float outv[64];
#pragma unroll
  for (int j = 0; j < 64; ++j) outv[j] = 0.f;
  for (int k = 0; k < CH; ++k) {
    const float a = A[row * CH + k];
    const float* brow = Bm + k * CH + c0;
#pragma unroll 8
    for (int j = 0; j < 64; ++j) outv[j] += a * brow[j];
  }
#pragma unroll 8
  for (int j = 0; j < 64; ++j) Cm[row * CH + c0 + j] = outv[j];
}

// ---------------------------------------------------------------------------
// Kernel 3: reduce Gram partials -> Sigma, trace-normalize, 5 Newton
// iterations, M = rot @ (P * sqrt(rTr)), bias = M @ mean.
// Single workgroup; 4 x 64KB matrices in dynamic LDS (256 KB of 320 KB WGP LDS).
// ---------------------------------------------------------------------------
__global__ void __launch_bounds__(256)
cw_newton(const float* __restrict__ gramPart, const float* __restrict__ mean,
          const float* __restrict__ rot, float* __restrict__ Mout,
          float* __restrict__ bias) {
  extern __shared__ float smem[];
  float* S  = smem;                 // SigmaN
  float* P  = smem + 1 * CH * CH;
  float* T1 = smem + 2 * CH * CH;
  float* T2 = smem + 3 * CH * CH;
  __shared__ float sTrInv;
  const int tid = threadIdx.x;

  // Sigma = (G - m * mu mu^T)/m + eps*I  (fixed-order partial reduction)
  for (int e = tid; e < CH * CH; e += 256) {
    float g = 0.f;
    for (int s = 0; s < BATCH; ++s) g += gramPart[(size_t)s * CH * CH + e];
    const int r = e >> 7, c = e & 127;
    float sig = (g - MTOT * mean[r] * mean[c]) / MTOT;
    if (r == c) sig += EPSR;
    S[e] = sig;
  }
  __syncthreads();
  if (tid == 0) {
    float tr = 0.f;
    for (int i = 0; i < CH; ++i) tr += S[i * CH + i];
    sTrInv = 1.0f / tr;
  }
  __syncthreads();
  const float rTr = sTrInv;
  for (int e = tid; e < CH * CH; e += 256) {
    S[e] *= rTr;
    const int r = e >> 7, c = e & 127;
    P[e] = (r == c) ? 1.f : 0.f;
  }
  __syncthreads();

  // Newton-Schulz: P = 0.5 * (3P - P^3 SigmaN)
  for (int it = 0; it < T_NEWTON; ++it) {
    mm128(T1, P, P, tid);  __syncthreads();
    mm128(T2, T1, P, tid); __syncthreads();
    mm128(T1, T2, S, tid); __syncthreads();
    for (int e = tid; e < CH * CH; e += 256) P[e] = 1.5f * P[e] - 0.5f * T1[e];
    __syncthreads();
  }

  // T2 = rot @ P ;  M = T2 * sqrt(rTr) ;  bias = M @ mean
  mm128(T2, rot, P, tid);
  __syncthreads();
  const float sc = sqrtf(rTr);
  for (int e = tid; e < CH * CH; e += 256) Mout[e] = T2[e] * sc;
  if (tid < CH) {
    float s = 0.f;
    for (int c = 0; c < CH; ++c) s += T2[tid * CH + c] * mean[c];
    bias[tid] = s * sc;
  }
}

// ---------------------------------------------------------------------------
// Kernel 4: out[b,d,:] = M @ x[b,:,:] - bias[d].
// Grid (7, 64): block = one batch slab, 7 chunks of 64 spatial positions.
// M staged once per block in LDS; x slabs staged per chunk; 8 waves x 4 tiles.
// ---------------------------------------------------------------------------
constexpr int MPAD = CH + 4;   // 132: LDS row stride for M
__global__ void __launch_bounds__(256)
cw_apply(const float* __restrict__ x, const float* __restrict__ Mmat,
         const float* __restrict__ bias, float* __restrict__ out) {
  extern __shared__ float smem[];
  float* sM = smem;                         // [128][132]
  float* sX = smem + CH * MPAD;             // [128][68]
  float* sB = sX + CH * XPAD;               // [128]
  const int b    = blockIdx.y;
  const int tid  = threadIdx.x;
  const int wave = tid >> 5;
  const int lane = tid & 31;
  const int hi   = lane >> 4;
  const int l15  = lane & 15;

  // Stage M (4096 float4, 16 per thread) and bias.
  for (int i = tid; i < (CH * CH) / 4; i += 256) {
    const int e = i * 4;
    const int r = e >> 7, c = e & 127;
    const float4 v = *reinterpret_cast<const float4*>(Mmat + e);
    float* d = sM + r * MPAD + c;
    d[0] = v.x; d[1] = v.y; d[2] = v.z; d[3] = v.w;
  }
  if (tid < CH) sB[tid] = bias[tid];
  __syncthreads();

  const float* xb = x + (size_t)b * CH * HW;
  float*       ob = out + (size_t)b * CH * HW;

  for (int j = 0; j < 7; ++j) {
    const int hw0 = (blockIdx.x * 7 + j) * 64;
    // Stage 128 x 64 slab of x.
#pragma unroll
    for (int i = 0; i < 8; ++i) {
      const int idx = i * 256 + tid;
      const int c   = idx >> 4;
      const int k4  = (idx & 15) << 2;
      const float4 v = *reinterpret_cast<const float4*>(xb + (size_t)c * HW + hw0 + k4);
      float* d = sX + c * XPAD + k4;
      d[0] = v.x; d[1] = v.y; d[2] = v.z; d[3] = v.w;
    }
    __syncthreads();

    v8f acc[4] = {};
#pragma unroll 2
    for (int c0 = 0; c0 < CH; c0 += 4) {
      const int ks = c0 + hi * 2;
      const int ra = wave * 16 + l15;
      v2f a; a.x = sM[ra * MPAD + ks]; a.y = sM[ra * MPAD + ks + 1];
#pragma unroll
      for (int ht = 0; ht < 4; ++ht) {
        v2f bb;
        bb.x = sX[ks * XPAD + ht * 16 + l15];
        bb.y = sX[(ks + 1) * XPAD + ht * 16 + l15];
        acc[ht] = __builtin_amdgcn_wmma_f32_16x16x4_f32(
            false, a, false, bb, (short)0, acc[ht], false, false);
      }
    }

    // Store with fused bias subtract (lanes 0-15 write 16 consecutive floats).
#pragma unroll
    for (int ht = 0; ht < 4; ++ht) {
#pragma unroll
      for (int r = 0; r < 8; ++r) {
        const int dch = wave * 16 + r + hi * 8;
        const int hw  = hw0 + ht * 16 + l15;
        ob[(size_t)dch * HW + hw] = acc[ht][r] - sB[dch];
      }
    }
    __syncthreads();
  }
}

// ---------------------------------------------------------------------------
// Host launcher.
// ---------------------------------------------------------------------------
extern "C" void kernel_launch(void* const* d_in, const int* in_sizes, int n_in,
                              void* d_out, int out_size, void* d_ws, size_t ws_size,
                              hipStream_t stream) {
  const float* x   = (const float*)d_in[0];
  const float* rot = (const float*)d_in[1];
  float* out = (float*)d_out;
  float* ws  = (float*)d_ws;

  // Workspace layout (floats): ~4.3 MB total.
  float* gramPart = ws;                                  // 64 * 128 * 128
  float* meanPart = gramPart + (size_t)BATCH * CH * CH;  // 128 * 64
  float* mean     = meanPart + CH * BATCH;               // 128
  float* Mmat     = mean + CH;                           // 128 * 128
  float* bias     = Mmat + CH * CH;                      // 128

  cw_mean_partial<<<dim3(BATCH, CH), 256, 0, stream>>>(x, meanPart);
  cw_gram<<<BATCH, 256, 0, stream>>>(x, gramPart);
  cw_mean_finish<<<1, CH, 0, stream>>>(meanPart, mean);
  cw_newton<<<1, 256, 4 * CH * CH * sizeof(float), stream>>>(
      gramPart, mean, rot, Mmat, bias);
  cw_apply<<<dim3(7, BATCH), 256,
             (CH * MPAD + CH * XPAD + CH) * sizeof(float), stream>>>(
      x, Mmat, bias, out);
}